// LIFNeuron_43173011259416
// MI455X (gfx1250) — compile-verified
//
#include <hip/hip_runtime.h>

// LIF constants (match reference)
#define T_STEPS 64
#define TAU_INV 0.1f      // 1/tauMem
#define U_REST  (-75.0f)
#define REF_T   2.0f
#define RES_MEM 0.1f
#define THRESH  (-55.0f)

typedef __attribute__((ext_vector_type(4))) float float4v;

// One thread owns 4 consecutive neurons. State lives in VGPRs for all 64
// steps; the only traffic is the mandatory 1.08 GB history-store stream,
// emitted as coalesced global_store_b128 with non-temporal hints (write-once
// data, 5.6x the 192 MB L2 -> keep it out of the cache's working set).
__global__ __launch_bounds__(256) void lif_neuron_kernel(
    const float* __restrict__ x,
    const float* __restrict__ mem0,
    const float* __restrict__ ref0,
    float* __restrict__ out,
    long long n)  // total neurons (multiple of 4)
{
    long long tid = (long long)blockIdx.x * blockDim.x + threadIdx.x;
    long long i = tid * 4;
    if (i >= n) return;

    // Load inputs exactly once (global_load_b128 each).
    float4v xv  = *(const float4v*)(x    + i);
    float4v mem = *(const float4v*)(mem0 + i);
    float4v ref = *(const float4v*)(ref0 + i);

    float4v xr;
#pragma unroll
    for (int k = 0; k < 4; ++k) xr[k] = xv[k] * RES_MEM;  // invariant: x*resMem

    float* __restrict__ mem_hist = out;                              // [T, n]
    float* __restrict__ spk_hist = out + (long long)T_STEPS * n;     // [T, n]
    float* __restrict__ mem_fin  = out + 2LL * (long long)T_STEPS * n; // [n]

#pragma unroll 4
    for (int t = 0; t < T_STEPS; ++t) {
        float4v spk;
#pragma unroll
        for (int k = 0; k < 4; ++k) {
            // refMaskInv: 1 where refractory (counter > 0)
            float ra  = (ref[k] > 0.0f) ? 1.0f : 0.0f;
            // input blocked while refractory
            float inp = (ra > 0.0f) ? 0.0f : xr[k];
            // mem += (1/tau) * ( (uRest - mem) + x*R*(1-ra) )
            mem[k] = fmaf(TAU_INV, (U_REST - mem[k]) + inp, mem[k]);
            // spike when mem - thr >= 0
            float s = (mem[k] >= THRESH) ? 1.0f : 0.0f;
            spk[k] = s;
            // spiking neurons get refCounter = refTime
            ref[k] = (s > 0.0f) ? REF_T : ref[k];
            // subtract reset: mem -= s * thr
            mem[k] = fmaf(s, -THRESH, mem[k]);
            // decrement counter where refractory before this step's spike
            ref[k] = ref[k] - ra;
        }
        long long off = (long long)t * n + i;
        // Write-once history: non-temporal b128 stores.
        __builtin_nontemporal_store(mem, (float4v*)(mem_hist + off));
        __builtin_nontemporal_store(spk, (float4v*)(spk_hist + off));
    }

    // Final membrane state (small, regular store).
    *(float4v*)(mem_fin + i) = mem;
}

extern "C" void kernel_launch(void* const* d_in, const int* in_sizes, int n_in,
                              void* d_out, int out_size, void* d_ws, size_t ws_size,
                              hipStream_t stream) {
    const float* x    = (const float*)d_in[0];
    const float* mem0 = (const float*)d_in[1];
    const float* ref0 = (const float*)d_in[2];
    float* out = (float*)d_out;

    long long n = (long long)in_sizes[0];         // 2^21 neurons
    long long threads = (n + 3) / 4;              // 4 neurons per thread
    const int block = 256;                        // 8 wave32s per workgroup
    long long grid = (threads + block - 1) / block;

    lif_neuron_kernel<<<dim3((unsigned)grid), dim3(block), 0, stream>>>(
        x, mem0, ref0, out, n);
}